// Neighborlist_40295383171534
// MI455X (gfx1250) — compile-verified
//
#include <hip/hip_runtime.h>
#include <stdint.h>

// ---------------------------------------------------------------------------
// Neighborlist screening kernel for MI455X (gfx1250, wave32).
//
//   - indices + outputs: streamed once  -> non-temporal (TH_NT) b128 ops
//   - coords + mask:     3MB + 256KB    -> L2-resident, default (RT) gathers
//   - global_prefetch_b8 pre-stages the index streams (clamped, branchless)
// Four pairs per thread => all streamed traffic is full-width b128,
// fully coalesced across each 32-lane wave.
// ---------------------------------------------------------------------------

typedef __attribute__((ext_vector_type(2))) long long v2i64;
typedef __attribute__((ext_vector_type(4))) float     v4f;

#define NL_BLOCK 256
// prefetch distance in index elements (8B each): 256KB ahead per stream
#define NL_PF_DIST 32768LL

__global__ __launch_bounds__(NL_BLOCK) void
neighborlist_gfx1250_kernel(const float* __restrict__ coords,        // (N,3) f32
                            const long long* __restrict__ nbr,       // (2,P) i64
                            const unsigned char* __restrict__ mask,  // (N,)  bool
                            const int* __restrict__ cutoff_ptr,      // (1,)  i32
                            float* __restrict__ out,                 // [P | 3P | P]
                            long long P)
{
    const long long t  = (long long)blockIdx.x * NL_BLOCK + threadIdx.x;
    const long long p0 = t * 4;              // first pair handled by this thread
    if (p0 >= P) return;

    // cutoff: uniform scalar load (s_load_b32 + s_cvt_f32_i32)
    const float cut = (float)cutoff_ptr[0];

    // --- branchless prefetch of the streaming index data (global_prefetch_b8)
    {
        const long long pf = (p0 + NL_PF_DIST < P) ? (p0 + NL_PF_DIST) : (P - 1);
        __builtin_prefetch(nbr + pf, 0, 3);          // i-stream
        __builtin_prefetch(nbr + P + pf, 0, 3);      // j-stream
    }

    // --- streamed, non-temporal 2x16B index loads per stream (th:TH_LOAD_NT)
    const v2i64 iv0 = __builtin_nontemporal_load((const v2i64*)(nbr + p0));
    const v2i64 iv1 = __builtin_nontemporal_load((const v2i64*)(nbr + p0 + 2));
    const v2i64 jv0 = __builtin_nontemporal_load((const v2i64*)(nbr + P + p0));
    const v2i64 jv1 = __builtin_nontemporal_load((const v2i64*)(nbr + P + p0 + 2));

    const int ii[4] = { (int)iv0.x, (int)iv0.y, (int)iv1.x, (int)iv1.y };
    const int jj[4] = { (int)jv0.x, (int)jv0.y, (int)jv1.x, (int)jv1.y };

    float dist[4], dfx[4], dfy[4], dfz[4], valf[4];

#pragma unroll
    for (int k = 0; k < 4; ++k) {
        const int i = ii[k];
        const int j = jj[k];

        // L2-resident random gathers (regular temporal hint, b96)
        const float* ci = coords + (long long)i * 3;
        const float* cj = coords + (long long)j * 3;
        const float dx = ci[0] - cj[0];
        const float dy = ci[1] - cj[1];
        const float dz = ci[2] - cj[2];

        const float sq = dx * dx + dy * dy + dz * dz;
        const float d  = sqrtf(sq);          // sqrt(0) == 0, matches safe_norm fwd

        const bool dummy = (mask[i] | mask[j]) != 0;
        const bool ok    = (!dummy) && (d <= cut);

        dist[k] = ok ? d  : 0.0f;
        dfx[k]  = ok ? dx : 0.0f;
        dfy[k]  = ok ? dy : 0.0f;
        dfz[k]  = ok ? dz : 0.0f;
        valf[k] = ok ? 1.0f : 0.0f;
    }

    // --- streamed, non-temporal full-width b128 stores (th:TH_STORE_NT) ----
    // distances: out[0 .. P), 16B aligned (p0 = 4t)
    {
        v4f v; v.x = dist[0]; v.y = dist[1]; v.z = dist[2]; v.w = dist[3];
        __builtin_nontemporal_store(v, (v4f*)(out + p0));
    }
    // diff_vectors: out[P .. 4P), row-major (P,3); 12 consecutive floats here,
    // 48B stride per thread -> 16B aligned
    {
        float* db = out + P + p0 * 3;
        v4f a; a.x = dfx[0]; a.y = dfy[0]; a.z = dfz[0]; a.w = dfx[1];
        v4f b; b.x = dfy[1]; b.y = dfz[1]; b.z = dfx[2]; b.w = dfy[2];
        v4f c; c.x = dfz[2]; c.y = dfx[3]; c.z = dfy[3]; c.w = dfz[3];
        __builtin_nontemporal_store(a, (v4f*)(db + 0));
        __builtin_nontemporal_store(b, (v4f*)(db + 4));
        __builtin_nontemporal_store(c, (v4f*)(db + 8));
    }
    // valid (as 0.0/1.0 float): out[4P .. 5P), 16B aligned
    {
        v4f v; v.x = valf[0]; v.y = valf[1]; v.z = valf[2]; v.w = valf[3];
        __builtin_nontemporal_store(v, (v4f*)(out + 4 * P + p0));
    }
}

extern "C" void kernel_launch(void* const* d_in, const int* in_sizes, int n_in,
                              void* d_out, int out_size, void* d_ws, size_t ws_size,
                              hipStream_t stream)
{
    const float*         coords = (const float*)d_in[0];         // (1,N,3) f32
    const long long*     nbr    = (const long long*)d_in[1];     // (2,P)  i64
    const unsigned char* mask   = (const unsigned char*)d_in[2]; // (N,)   bool
    const int*           cutoff = (const int*)d_in[3];           // scalar i32
    float*               out    = (float*)d_out;                 // 5P floats

    const long long P = (long long)in_sizes[1] / 2;              // 8,388,608
    const long long nthreads = (P + 3) / 4;                      // 4 pairs/thread
    const int nblocks = (int)((nthreads + NL_BLOCK - 1) / NL_BLOCK);

    neighborlist_gfx1250_kernel<<<nblocks, NL_BLOCK, 0, stream>>>(
        coords, nbr, mask, cutoff, out, P);
}